// TVBallProj_5806795784696
// MI455X (gfx1250) — compile-verified
//
#include <hip/hip_runtime.h>
#include <math.h>
#include <stdint.h>

// ---------------------------------------------------------------------------
// Accelerated TV-ball projection, MI455X (gfx1250) persistent-LDS solver.
//   - 1 workgroup / batch image; u, y, G resident in the 320KB WGP LDS
//   - xi lives in VGPRs laid out to match WMMA D-fragments exactly
//   - grad expressed as bidiagonal-matrix products executed with
//     v_wmma_f32_16x16x4_f32 (entries 0/±1, G pre-scaled by -mu/2 where
//     needed -> bit-exact fp32), straight-line (EXEC all-ones at every WMMA)
//   - coefficient/address rematerialization forced via empty volatile asm
//     (recompute beats hoist+scratch-spill in front of the WMMA chain)
//   - L1-ball projection via Michelot fixed point; scalar control flow
//   - outputs drained with global_store_async_from_lds_b128 / s_wait_asynccnt
// ---------------------------------------------------------------------------

#define N        128
#define NP       (N * N)            // 16384 pixels / image
#define NTHREADS 1024               // 32 wave32s
#define PPT      (NP / NTHREADS)    // 16 pixels / thread
#define NITERS   200
#define MU_C     0.25f

// Dynamic LDS partition (floats): uL[2NP] | yL[2NP] | G[NP]  == 320KB exactly.
// Reduction scratch (128 floats) aliases yL[0:128] (dead during reductions).
#define OFF_U    0
#define OFF_Y    (2 * NP)
#define OFF_G    (4 * NP)
#define SMEM_FLOATS (5 * NP)

#define OPAQUE(x) asm volatile("" : "+v"(x))

static_assert(NTHREADS / 32 == 32, "reduction stage assumes exactly 32 waves");

typedef __attribute__((ext_vector_type(2))) float v2f;
typedef __attribute__((ext_vector_type(8))) float v8f;

__device__ __forceinline__ float rdff(float x) {
  return __int_as_float(__builtin_amdgcn_readfirstlane(__float_as_int(x)));
}

// ---- block-wide sum of 4 floats; results returned as wave-uniform scalars --
__device__ __forceinline__ void block_reduce4(float& a, float& b, float& c, float& d,
                                              float* red) {
  #pragma unroll
  for (int m = 16; m >= 1; m >>= 1) {
    a += __shfl_xor(a, m, 32);
    b += __shfl_xor(b, m, 32);
    c += __shfl_xor(c, m, 32);
    d += __shfl_xor(d, m, 32);
  }
  const int lane = threadIdx.x & 31;
  const int wid  = threadIdx.x >> 5;
  if (lane == 0) {
    red[wid * 4 + 0] = a; red[wid * 4 + 1] = b;
    red[wid * 4 + 2] = c; red[wid * 4 + 3] = d;
  }
  __syncthreads();
  // every wave redundantly reduces the 32 partials (no divergent region)
  a = red[lane * 4 + 0]; b = red[lane * 4 + 1];
  c = red[lane * 4 + 2]; d = red[lane * 4 + 3];
  #pragma unroll
  for (int m = 16; m >= 1; m >>= 1) {
    a += __shfl_xor(a, m, 32);
    b += __shfl_xor(b, m, 32);
    c += __shfl_xor(c, m, 32);
    d += __shfl_xor(d, m, 32);
  }
  __syncthreads();                  // protect red[] before next reuse
  a = rdff(a); b = rdff(b); c = rdff(c); d = rdff(d);   // scalarize
}

// ---- Michelot L1-ball thresholds for both components at once --------------
template <bool SUBXI>
__device__ __forceinline__ void l1_thresholds(const float* uL, const float* xi,
                                              const int* pbase, float R, float* red,
                                              float& thX, float& thY) {
  float Sx = 0.f, Sy = 0.f, z0 = 0.f, z1 = 0.f;
  #pragma unroll
  for (int k = 0; k < PPT; ++k) {
    const int p = pbase[k >> 3] + (k & 7) * N;
    float wx = uL[2 * p + 0];
    float wy = uL[2 * p + 1];
    if constexpr (SUBXI) { wx -= xi[2 * k + 0]; wy -= xi[2 * k + 1]; }
    Sx += fabsf(wx); Sy += fabsf(wy);
  }
  block_reduce4(Sx, Sy, z0, z1, red);   // Sx,Sy now scalar

  bool aX = Sx > R, aY = Sy > R;        // scalar: sum|w| <= R => shrink == 0
  thX = aX ? (Sx - R) * (1.0f / NP) : 0.0f;
  thY = aY ? (Sy - R) * (1.0f / NP) : 0.0f;

  for (int it = 0; it < 24 && (aX || aY); ++it) {   // scalar loop control
    float sx = 0.f, kx = 0.f, sy = 0.f, ky = 0.f;
    #pragma unroll
    for (int k = 0; k < PPT; ++k) {
      const int p = pbase[k >> 3] + (k & 7) * N;
      float wx = uL[2 * p + 0];
      float wy = uL[2 * p + 1];
      if constexpr (SUBXI) { wx -= xi[2 * k + 0]; wy -= xi[2 * k + 1]; }
      wx = fabsf(wx); wy = fabsf(wy);
      if (wx > thX) { sx += wx; kx += 1.0f; }
      if (wy > thY) { sy += wy; ky += 1.0f; }
    }
    block_reduce4(sx, kx, sy, ky, red); // scalars after call
    if (aX) { const float nt = (kx > 0.f) ? (sx - R) / kx : thX;
              if (nt == thX) aX = false; else thX = nt; }
    if (aY) { const float nt = (ky > 0.f) ? (sy - R) / ky : thY;
              if (nt == thY) aY = false; else thY = nt; }
  }
}

// ---- negative-adjoint divergence of an interleaved LDS field --------------
__device__ __forceinline__ float divF(const float* F, int r, int c) {
  const int p = r * N + c;
  float dx, dy;
  if (r == 0)          dx =  F[2 * p + 0];
  else if (r == N - 1) dx = -F[2 * (p - N) + 0];
  else                 dx =  F[2 * p + 0] - F[2 * (p - N) + 0];
  if (c == 0)          dy =  F[2 * p + 1];
  else if (c == N - 1) dy = -F[2 * (p - 1) + 1];
  else                 dy =  F[2 * p + 1] - F[2 * (p - 1) + 1];
  return dx + dy;
}

// Dg[i][k]: forward-difference matrix (grad), zero last row. Entries {0,+-1}.
__device__ __forceinline__ float dgcoef(int i, int k) {
  if (i >= N - 1) return 0.f;
  if (k == i + 1) return 1.f;
  if (k == i)     return -1.f;
  return 0.f;
}

// P = Dg*G + C  for one 16x16 tile at (i0,j0).  Always 5 K-chunks, fully
// unrolled (straight-line: EXEC stays all-ones).  For edge tiles the 5th
// chunk's Dg coefficients are exactly 0, so G indices are just clamped.
__device__ __forceinline__ v8f wmma_gradx(const float* G, int i0, int j0,
                                          int ln, int hl, v8f acc) {
  #pragma unroll
  for (int c4 = 0; c4 < 5; ++c4) {
    const int kk = i0 + 4 * c4 + 2 * hl;
    const int ia = i0 + ln;
    v2f a, bf;
    a.x  = dgcoef(ia, kk);
    a.y  = dgcoef(ia, kk + 1);
    bf.x = G[(kk & (N - 1)) * N + j0 + ln];        // B rows (VGPR0: K=0/2; VGPR1: K=1/3)
    bf.y = G[((kk + 1) & (N - 1)) * N + j0 + ln];
    acc = __builtin_amdgcn_wmma_f32_16x16x4_f32(false, a, false, bf,
                                                (short)0, acc, false, false);
  }
  return acc;
}

// P = G*Dg^T + C  for one 16x16 tile at (i0,j0)
__device__ __forceinline__ v8f wmma_grady(const float* G, int i0, int j0,
                                          int ln, int hl, v8f acc) {
  #pragma unroll
  for (int c4 = 0; c4 < 5; ++c4) {
    const int kk = j0 + 4 * c4 + 2 * hl;
    v2f a, bf;
    a.x  = G[(i0 + ln) * N + (kk & (N - 1))];      // A[m][*] = G row i0+m
    a.y  = G[(i0 + ln) * N + ((kk + 1) & (N - 1))];
    bf.x = dgcoef(j0 + ln, kk);                    // B[k][n] = Dg[j0+n][k]
    bf.y = dgcoef(j0 + ln, kk + 1);
    acc = __builtin_amdgcn_wmma_f32_16x16x4_f32(false, a, false, bf,
                                                (short)0, acc, false, false);
  }
  return acc;
}

// ---- async LDS -> global drain (CDNA5 ASYNCcnt path) ----------------------
__device__ __forceinline__ void async_store_b128(const float* gptr, const float* ldsptr) {
  const unsigned long long ga  = (unsigned long long)(uintptr_t)gptr;
  const unsigned           lds = (unsigned)(uintptr_t)(const void*)ldsptr;
  asm volatile("global_store_async_from_lds_b128 %0, %1, off"
               :: "v"(ga), "v"(lds) : "memory");
}

__global__ void __launch_bounds__(NTHREADS, 1)
tv_agm_kernel(const float* __restrict__ Mfull, const float* __restrict__ tv0,
              float* __restrict__ outM, float* __restrict__ outU) {
  extern __shared__ float smem[];
  float* uL  = smem + OFF_U;
  float* yL  = smem + OFF_Y;
  float* Gf  = smem + OFF_G;
  float* red = yL;                      // aliases yL[0:128]; dead there then

  const int   t  = threadIdx.x;
  const int   b  = blockIdx.x;
  const float* Mg = Mfull + b * NP;
  const float r0  = tv0[0];

  // WMMA-fragment-derived pixel ownership: wave w owns tiles 2w, 2w+1
  // (both tiles share the same row block -> same i0).
  const int ln = t & 15;                // column within tile / A-row m
  const int hl = (t >> 4) & 1;          // lane >= 16 -> +8 rows (D layout)
  int tIs[2], tJs[2], pbase[2];
  #pragma unroll
  for (int tl = 0; tl < 2; ++tl) {
    const int tidx = __builtin_amdgcn_readfirstlane(2 * (t >> 5) + tl); // 0..63
    tIs[tl] = (tidx >> 3) * 16;         // scalar tile origins
    tJs[tl] = (tidx & 7) * 16;
    pbase[tl] = (tIs[tl] + 8 * hl) * N + tJs[tl] + ln;  // D-frag k=0 pixel
  }

  float xi[2 * PPT];
  #pragma unroll
  for (int k = 0; k < 2 * PPT; ++k) xi[k] = 0.f;

  for (int k = 0; k < PPT; ++k) {       // coalesced init + cache warmup
    const int p = t + k * NTHREADS;
    uL[2 * p + 0] = 0.f; uL[2 * p + 1] = 0.f;
    yL[2 * p + 0] = 0.f; yL[2 * p + 1] = 0.f;
    __builtin_prefetch(Mg + p, 0, 1);   // global_prefetch_b8
  }
  __syncthreads();

  float A = 0.f;
  for (int it = 0; it < NITERS; ++it) {
    const float a = 0.5f * (MU_C + sqrtf(MU_C * MU_C + 4.0f * MU_C * A));

    // Phase 1: thresholds for v = shrink(u - xi, r0*A)
    float thX, thY;
    l1_thresholds<true>(uL, xi, pbase, r0 * A, red, thX, thY);

    // Phase 2: y = (A*u + a*v)/(A+a)   (own pixels)
    const float inv = 1.0f / (A + a);
    #pragma unroll
    for (int k = 0; k < PPT; ++k) {
      const int p = pbase[k >> 3] + (k & 7) * N;
      const float ux = uL[2 * p + 0], uy = uL[2 * p + 1];
      const float wx = ux - xi[2 * k + 0], wy = uy - xi[2 * k + 1];
      const float vx = copysignf(fminf(fabsf(wx), thX), wx);
      const float vy = copysignf(fminf(fabsf(wy), thY), wy);
      yL[2 * p + 0] = (A * ux + a * vx) * inv;
      yL[2 * p + 1] = (A * uy + a * vy) * inv;
    }
    __syncthreads();

    // Phase 3a: G = -(mu/2)*(M - div(y))   (pre-scaled: exact *2^-3; lets
    // both WMMA phases share the identical {0,±1} coefficient fragments)
    for (int k = 0; k < PPT; ++k) {
      const int p = t + k * NTHREADS;
      Gf[p] = (-0.5f * MU_C) * (Mg[p] - divF(yL, p >> 7, p & (N - 1)));
    }
    __syncthreads();

    // Phase 3b: w2 = y + grad(G) via WMMA, folded C = y  -> uL
    {
      int lnq = ln, hlq = hl;
      OPAQUE(lnq); OPAQUE(hlq);       // force per-iteration remat (no spills)
      #pragma unroll
      for (int tl = 0; tl < 2; ++tl) {
        const int i0 = tIs[tl], j0 = tJs[tl];
        {
          v8f cx;
          #pragma unroll
          for (int k = 0; k < 8; ++k) cx[k] = yL[2 * (pbase[tl] + k * N) + 0];
          cx = wmma_gradx(Gf, i0, j0, lnq, hlq, cx);
          #pragma unroll
          for (int k = 0; k < 8; ++k) uL[2 * (pbase[tl] + k * N) + 0] = cx[k];
        }
        {
          v8f cy;
          #pragma unroll
          for (int k = 0; k < 8; ++k) cy[k] = yL[2 * (pbase[tl] + k * N) + 1];
          cy = wmma_grady(Gf, i0, j0, lnq, hlq, cy);
          #pragma unroll
          for (int k = 0; k < 8; ++k) uL[2 * (pbase[tl] + k * N) + 1] = cy[k];
        }
      }
    }
    __syncthreads();

    // Phase 4: u = shrink(w2, r0*mu/2)  (in place, own pixels)
    l1_thresholds<false>(uL, xi, pbase, r0 * MU_C * 0.5f, red, thX, thY);
    #pragma unroll
    for (int k = 0; k < PPT; ++k) {
      const int p = pbase[k >> 3] + (k & 7) * N;
      const float wx = uL[2 * p + 0], wy = uL[2 * p + 1];
      uL[2 * p + 0] = copysignf(fminf(fabsf(wx), thX), wx);
      uL[2 * p + 1] = copysignf(fminf(fabsf(wy), thY), wy);
    }
    __syncthreads();

    // Phase 5a: G = Mnew = M - div(u)   (unscaled)
    for (int k = 0; k < PPT; ++k) {
      const int p = t + k * NTHREADS;
      Gf[p] = Mg[p] - divF(uL, p >> 7, p & (N - 1));
    }
    __syncthreads();

    // Phase 5b: xi += a * grad(Mnew) via WMMA (D-fragment == xi layout)
    {
      int lnq = ln, hlq = hl;
      OPAQUE(lnq); OPAQUE(hlq);
      #pragma unroll
      for (int tl = 0; tl < 2; ++tl) {
        const int i0 = tIs[tl], j0 = tJs[tl];
        {
          v8f gx = {};
          gx = wmma_gradx(Gf, i0, j0, lnq, hlq, gx);
          #pragma unroll
          for (int k = 0; k < 8; ++k) xi[2 * (8 * tl + k) + 0] += a * gx[k];
        }
        {
          v8f gy = {};
          gy = wmma_grady(Gf, i0, j0, lnq, hlq, gy);
          #pragma unroll
          for (int k = 0; k < 8; ++k) xi[2 * (8 * tl + k) + 1] += a * gy[k];
        }
      }
    }
    A += a;
    __syncthreads();
  }

  // Finalize: M1 = M - div(u) staged into (dead) y buffer
  for (int k = 0; k < PPT; ++k) {
    const int p = t + k * NTHREADS;
    yL[p] = Mg[p] - divF(uL, p >> 7, p & (N - 1));
  }
  __syncthreads();

  // Drain LDS -> HBM with async stores (contiguous 16B chunks per thread)
  {
    const float* gM = outM + b * NP;
    #pragma unroll
    for (int j = 0; j < 4; ++j) {
      const int idx = t * 16 + j * 4;       // covers [0, NP)
      async_store_b128(gM + idx, &yL[idx]);
    }
    const float* gU = outU + b * 2 * NP;    // [N,N,2] == interleaved uL layout
    #pragma unroll
    for (int j = 0; j < 8; ++j) {
      const int idx = t * 32 + j * 4;       // covers [0, 2*NP)
      async_store_b128(gU + idx, &uL[idx]);
    }
  }
  asm volatile("s_wait_asynccnt 0x0" ::: "memory");
}

extern "C" void kernel_launch(void* const* d_in, const int* in_sizes, int n_in,
                              void* d_out, int out_size, void* d_ws, size_t ws_size,
                              hipStream_t stream) {
  (void)in_sizes; (void)n_in; (void)out_size; (void)d_ws; (void)ws_size;
  const float* M   = (const float*)d_in[0];
  const float* tv0 = (const float*)d_in[1];
  float* outM = (float*)d_out;             // [8,1,128,128]
  float* outU = (float*)d_out + 8 * NP;    // [8,1,128,128,2]

  const size_t smem = SMEM_FLOATS * sizeof(float);   // 320 KB WGP LDS
  (void)hipFuncSetAttribute((const void*)tv_agm_kernel,
                            hipFuncAttributeMaxDynamicSharedMemorySize, (int)smem);
  tv_agm_kernel<<<8, NTHREADS, smem, stream>>>(M, tv0, outM, outU);
}